// FilterDetection_6055903887866
// MI455X (gfx1250) — compile-verified
//
#include <hip/hip_runtime.h>
#include <math.h>

// ---------------------------------------------------------------------------
// FilterDetection for MI455X (gfx1250, wave32).
//  kernel 1 (nms_kernel, grid=B*80): per (image,class)
//    - gather 2000 class scores, bitonic-sort 2048 packed u64 keys in LDS
//    - decode only the top-500 boxes (delta2bbox) into LDS
//    - 16x16-tiled pairwise IoU: area_i+area_j via v_wmma_f32_16x16x4_f32
//      (exact f32 rank-2 update on the matrix pipe, co-executes with VALU
//      min/max), predicate packed with ballot_w32 -> ds_or bitmask matrix
//    - greedy NMS = 500-step bitmask OR scan on wave 0
//  kernel 2 (topk_kernel, grid=B): top-100 of 80x512 scratch scores.
// ---------------------------------------------------------------------------

typedef __attribute__((ext_vector_type(2))) float v2f;
typedef __attribute__((ext_vector_type(8))) float v8f;

#define Bimg 16
#define Nprop 2000
#define Ncls 81
#define FG 80
#define PRE 500
#define PREP 512
#define TOPP 100
#define IOU_THRc 0.5f
#define SCORE_THRc 0.05f
#define MAXR 4.1351665f   // |log(16/1000)|

__device__ __forceinline__ unsigned ordKey(float f) {
  unsigned u = __float_as_uint(f);
  return (u & 0x80000000u) ? ~u : (u | 0x80000000u);
}
__device__ __forceinline__ float invOrd(unsigned k) {
  unsigned u = (k & 0x80000000u) ? (k & 0x7FFFFFFFu) : ~k;
  return __uint_as_float(u);
}

__global__ __launch_bounds__(256) void nms_kernel(
    const float* __restrict__ logits, const float* __restrict__ regress,
    const float* __restrict__ props, float* __restrict__ wsScores,
    float* __restrict__ wsBoxes) {
  const int blk = blockIdx.x;
  const int img = blk / FG;
  const int cls = blk % FG;          // fg class index; real class = cls+1
  const int tid = threadIdx.x;

  __shared__ union {
    unsigned long long keys[2048];   // 16 KB (sort phase)
    unsigned int sup[PREP][16];      // 32 KB (suppression bitmask phase)
  } u;
  __shared__ float X1[PREP], Y1[PREP], X2[PREP], Y2[PREP], AR[PREP], SC[PREP];
  __shared__ unsigned int removed[16];

  // ---- 1) gather packed sort keys: (ordered score | ~index) ----
  for (int i = tid; i < 2048; i += 256) {
    unsigned long long key = 0ull;
    if (i < Nprop) {
      float s = logits[((size_t)img * Nprop + i) * Ncls + (cls + 1)];
      key = ((unsigned long long)ordKey(s) << 32) |
            (unsigned)(0xFFFFFFFFu - (unsigned)i);
    }
    u.keys[i] = key;
  }
  __syncthreads();

  // ---- 2) bitonic sort, descending (ties -> lower original index) ----
  for (int k = 2; k <= 2048; k <<= 1) {
    for (int j = k >> 1; j > 0; j >>= 1) {
      for (int i = tid; i < 2048; i += 256) {
        int l = i ^ j;
        if (l > i) {
          unsigned long long a = u.keys[i], b = u.keys[l];
          bool sw = ((i & k) == 0) ? (a < b) : (a > b);
          if (sw) { u.keys[i] = b; u.keys[l] = a; }
        }
      }
      __syncthreads();
    }
  }

  // ---- 3) decode only the top-500 boxes (delta2bbox) into LDS ----
  for (int r = tid; r < PREP; r += 256) {
    float x1 = 0.f, y1 = 0.f, x2 = 0.f, y2 = 0.f, s = 0.f;
    if (r < PRE) {
      unsigned long long key = u.keys[r];
      unsigned n = 0xFFFFFFFFu - (unsigned)key;
      s = invOrd((unsigned)(key >> 32));
      const float* p = props + ((size_t)img * Nprop + n) * 4;
      const float* d = regress + ((size_t)img * Nprop + n) * (size_t)(Ncls * 4)
                       + (size_t)(cls + 1) * 4;
      float dx = d[0] * 0.1f, dy = d[1] * 0.1f;
      float dw = fminf(fmaxf(d[2] * 0.2f, -MAXR), MAXR);
      float dh = fminf(fmaxf(d[3] * 0.2f, -MAXR), MAXR);
      float pw = p[2] - p[0], ph = p[3] - p[1];
      float cx = p[0] + 0.5f * pw + dx * pw;
      float cy = p[1] + 0.5f * ph + dy * ph;
      float w = pw * expf(dw), h = ph * expf(dh);
      x1 = fminf(fmaxf(cx - 0.5f * w, 0.f), 1.f);
      y1 = fminf(fmaxf(cy - 0.5f * h, 0.f), 1.f);
      x2 = fminf(fmaxf(cx + 0.5f * w, 0.f), 1.f);
      y2 = fminf(fmaxf(cy + 0.5f * h, 0.f), 1.f);
    }
    X1[r] = x1; Y1[r] = y1; X2[r] = x2; Y2[r] = y2;
    AR[r] = (x2 - x1) * (y2 - y1);
    SC[r] = s;
  }
  __syncthreads();

  // ---- 4) zero suppression bitmask (reuses sort-key LDS) ----
  for (int i = tid; i < PREP * 16; i += 256) ((unsigned*)u.sup)[i] = 0u;
  __syncthreads();

  // ---- 5) tiled pairwise IoU; area sums via WMMA on the matrix pipe ----
  const int wv = tid >> 5, lane = tid & 31;
  const int n_ = lane & 15, half = lane >> 4;
  int tcount = 0;
  for (int ti = 0; ti < 32; ++ti) {
    for (int tj = ti; tj < 32; ++tj) {
      if ((tcount++ & 7) != wv) continue;       // 528 tiles, 66 per wave
      const int i0 = ti * 16, j0 = tj * 16;
      // A(16x4): row m = [area_i, 1, 0, 0]; lanes 0-15 hold K=0,1
      // B(4x16): col n = [1, area_j, 0, 0]; lanes 0-15 hold K=0,1 rows
      float arI = AR[i0 + n_], arJ = AR[j0 + n_];
      bool lo = (half == 0);
      v2f a = { lo ? arI : 0.f, lo ? 1.f : 0.f };
      v2f b = { lo ? 1.f : 0.f, lo ? arJ : 0.f };
      v8f c = {0.f, 0.f, 0.f, 0.f, 0.f, 0.f, 0.f, 0.f};
      v8f dsum = __builtin_amdgcn_wmma_f32_16x16x4_f32(
          false, a, false, b, (short)0, c, false, false);
      // D layout: VGPR r -> (M = r + 8*half, N = n_)
      const int j = j0 + n_;
      const float cx1 = X1[j], cy1 = Y1[j], cx2 = X2[j], cy2 = Y2[j];
#pragma unroll
      for (int r = 0; r < 8; ++r) {
        const int i = i0 + r + half * 8;
        float iw = fminf(X2[i], cx2) - fmaxf(X1[i], cx1);
        float ih = fminf(Y2[i], cy2) - fmaxf(Y1[i], cy1);
        float inter = fmaxf(iw, 0.f) * fmaxf(ih, 0.f);
        float uni = fmaxf(dsum[r] - inter, 1e-9f);   // area_i+area_j-inter
        bool flag = (inter > IOU_THRc * uni) && (j > i);
        unsigned bal = __builtin_amdgcn_ballot_w32(flag);
        if (lane == 0) {
          unsigned sh = (unsigned)(j0 & 16);
          atomicOr(&u.sup[i0 + r][j0 >> 5], (bal & 0xFFFFu) << sh);
          atomicOr(&u.sup[i0 + 8 + r][j0 >> 5], (bal >> 16) << sh);
        }
      }
    }
  }
  __syncthreads();

  // ---- 6) greedy scan: wave 0, lane w owns removed-word w ----
  if (tid < 32) {
    unsigned rem = 0u;
    for (int i = 0; i < PRE; ++i) {
      bool bit = (tid == (i >> 5)) && ((rem >> (i & 31)) & 1u);
      unsigned bl = __builtin_amdgcn_ballot_w32(bit);
      if (bl == 0u) {                       // i survives: absorb its row
        if (tid < 16) rem |= u.sup[i][tid];
      }
    }
    if (tid < 16) removed[tid] = rem;
  }
  __syncthreads();

  // ---- 7) emit per-(image,class) scores (zeroed if suppressed) + boxes ----
  float* sOut = wsScores + (size_t)blk * PREP;
  float* bOut = wsBoxes + (size_t)blk * PREP * 4;
  for (int i = tid; i < PREP; i += 256) {
    float outS;
    if (i < PRE) {
      bool kept = ((removed[i >> 5] >> (i & 31)) & 1u) == 0u;
      float s = SC[i];
      outS = (kept && s > SCORE_THRc) ? s : 0.f;
    } else {
      outS = -1.f;                          // padding: below any real score
    }
    sOut[i] = outS;
    bOut[i * 4 + 0] = X1[i]; bOut[i * 4 + 1] = Y1[i];
    bOut[i * 4 + 2] = X2[i]; bOut[i * 4 + 3] = Y2[i];
  }
}

__global__ __launch_bounds__(256) void topk_kernel(
    float* __restrict__ wsScores, const float* __restrict__ wsBoxes,
    float* __restrict__ outBoxes, float* __restrict__ outScores,
    int* __restrict__ outLabels) {
  const int img = blockIdx.x, tid = threadIdx.x;
  const int M = FG * PREP;                  // 40960
  __shared__ float sv[256];
  __shared__ int si[256];
  float* sc = wsScores + (size_t)img * M;
  const float* bx = wsBoxes + (size_t)img * M * 4;
  for (int k = 0; k < TOPP; ++k) {
    float bv = -3.4e38f; int bi = 0x7FFFFFFF;
    for (int i = tid; i < M; i += 256) {
      float v = sc[i];
      if (v > bv || (v == bv && i < bi)) { bv = v; bi = i; }
    }
    sv[tid] = bv; si[tid] = bi;
    __syncthreads();
    for (int s = 128; s > 0; s >>= 1) {
      if (tid < s) {
        float ov = sv[tid + s]; int oi = si[tid + s];
        if (ov > sv[tid] || (ov == sv[tid] && oi < si[tid])) {
          sv[tid] = ov; si[tid] = oi;
        }
      }
      __syncthreads();
    }
    if (tid == 0) {
      int sel = si[0];
      int o = img * TOPP + k;
      outScores[o] = sv[0];
      outBoxes[o * 4 + 0] = bx[(size_t)sel * 4 + 0];
      outBoxes[o * 4 + 1] = bx[(size_t)sel * 4 + 1];
      outBoxes[o * 4 + 2] = bx[(size_t)sel * 4 + 2];
      outBoxes[o * 4 + 3] = bx[(size_t)sel * 4 + 3];
      outLabels[o] = (sel >> 9) + 1;        // class index / 512, +1 for fg
      sc[sel] = -3.4e38f;                   // exclude from later rounds
    }
    __syncthreads();
  }
}

extern "C" void kernel_launch(void* const* d_in, const int* in_sizes, int n_in,
                              void* d_out, int out_size, void* d_ws, size_t ws_size,
                              hipStream_t stream) {
  const float* logits = (const float*)d_in[0];   // (16,2000,81)
  const float* regress = (const float*)d_in[1];  // (16,2000,324)
  const float* props = (const float*)d_in[2];    // (16,2000,4)

  float* wsScores = (float*)d_ws;                              // 16*80*512
  float* wsBoxes = wsScores + (size_t)Bimg * FG * PREP;        // *4

  float* outBoxes = (float*)d_out;                             // 16*100*4
  float* outScores = outBoxes + (size_t)Bimg * TOPP * 4;       // 16*100
  int* outLabels = (int*)(outScores + (size_t)Bimg * TOPP);    // 16*100

  nms_kernel<<<Bimg * FG, 256, 0, stream>>>(logits, regress, props,
                                            wsScores, wsBoxes);
  topk_kernel<<<Bimg, 256, 0, stream>>>(wsScores, wsBoxes,
                                        outBoxes, outScores, outLabels);
}